// PostProcessor_66657892434357
// MI455X (gfx1250) — compile-verified
//
#include <hip/hip_runtime.h>
#include <hip/hip_bf16.h>

// ---------------- problem constants ----------------
#define N_BOXES     262144
#define NUM_CLASSES 80
#define ROW         85          // 4 box coords + 1 conf + 80 class probs
#define SCORE_TH    0.6f
#define IOU_TH      0.5f
#define MAX_SEL     300

// ---------------- phase 1 config ----------------
#define P1_THREADS  256
#define P1_ROWS     256         // rows (boxes) staged per block via TDM

// ---------------- phase 2 config ----------------
#define NWG         64          // persistent workgroups (must all be resident)
#define P2_THREADS  256
#define CHUNK       (N_BOXES / NWG)        // 4096 boxes per block
#define PER_T       (CHUNK / P2_THREADS)   // 16 boxes per thread
#define NWAVES      (P2_THREADS / 32)      // 8 waves (wave32)

typedef unsigned int v4u __attribute__((ext_vector_type(4)));
typedef int          v8i __attribute__((ext_vector_type(8)));
typedef int          v4i __attribute__((ext_vector_type(4)));

// =====================================================================
// Phase 1: scores/classes/threshold + packed box copy.
// Each block DMAs a 256x85 f32 tile into LDS via the Tensor Data Mover,
// waits on TENSORcnt, then computes per-row conf*prob max/argmax.
// =====================================================================
__global__ __launch_bounds__(P1_THREADS)
void nms_phase1(const float* __restrict__ in,
                float* __restrict__ live,
                int* __restrict__ cls,
                float4* __restrict__ boxesc)
{
    __shared__ float tile[P1_ROWS * ROW];   // 85 KB LDS (of 320 KB/WGP)

    const int t = threadIdx.x;
    const long long rowbase = (long long)blockIdx.x * P1_ROWS;

    // Wave 0 issues one TDM 2-D tile load: 85 x 256 f32, row stride 85.
    if (t < 32) {
        unsigned long long ga =
            (unsigned long long)(const void*)(in + rowbase * ROW);
        unsigned int ldsoff = (unsigned int)(unsigned long long)(const void*)&tile[0];

        // D# group 0: count=1 | lds_addr | global_addr[56:0] | type=2
        v4u g0 = { 1u,
                   ldsoff,
                   (unsigned int)ga,
                   (unsigned int)((ga >> 32) & 0x01FFFFFFu) | (2u << 30) };
        // D# group 1: data_size=4B, tensor_dim0=85, tensor_dim1=256,
        //             tile_dim0=85, tile_dim1=256, tensor_dim0_stride=85
        v8i g1 = { (int)(2u << 16),            // wg_mask=0, data_size=2 (4B)
                   (int)(85u << 16),           // tensor_dim0[15:0] << 16
                   (int)(256u << 16),          // dim0 hi=0 | tensor_dim1 lo
                   (int)(85u << 16),           // dim1 hi=0 | tile_dim0
                   256,                        // tile_dim1=256, tile_dim2=0
                   85,                         // tensor_dim0_stride lo32
                   0,                          // stride0 hi | stride1 lo
                   0 };                        // stride1 hi
        v4i g2 = { 1, 0, 0, 0 };               // dim2=1, unused beyond 2-D
        v4i g3 = { 0, 0, 0, 0 };
        v8i g4 = { 0, 0, 0, 0, 0, 0, 0, 0 };   // 6-arg toolchain: extra group (zeros)
        __builtin_amdgcn_tensor_load_to_lds(g0, g1, g2, g3, g4, 0);
        __builtin_amdgcn_s_wait_tensorcnt(0);  // s_wait_tensorcnt 0
    }
    __syncthreads();

    // Row per thread; LDS stride 85 is coprime with 64 banks -> conflict-free.
    const float* r = &tile[t * ROW];
    const float conf = r[4];
    float best = -1.0f;
    int bc = 0;
#pragma unroll
    for (int c = 0; c < NUM_CLASSES; ++c) {
        float s = conf * r[5 + c];
        if (s > best) { best = s; bc = c; }   // first max wins (jnp.argmax)
    }
    const long long gi = rowbase + t;
    live[gi]   = (best >= SCORE_TH) ? best : -1.0f;
    cls[gi]    = bc;
    boxesc[gi] = make_float4(r[0], r[1], r[2], r[3]);
}

// =====================================================================
// Phase 2 helpers
// =====================================================================
__device__ __forceinline__
void grid_barrier(unsigned int* cnt, unsigned int target)
{
    __syncthreads();
    if (threadIdx.x == 0) {
        __threadfence();                       // release candidate writes
        atomicAdd(cnt, 1u);
        while (atomicAdd(cnt, 0u) < target) {
            __builtin_amdgcn_s_sleep(1);       // polite L2 spin
        }
        __threadfence();                       // acquire (invalidate L0)
    }
    // HW cluster barrier: architectural NOP unless cluster-dispatched;
    // becomes the fast path under a future cluster launch.
    __builtin_amdgcn_s_cluster_barrier();
    __syncthreads();
}

__device__ __forceinline__
void block_argmax_to_cand(const float* s_sc, int base,
                          float* w_s, int* w_i,
                          float* cand_s, int* cand_i)
{
    const int t = threadIdx.x;
    float bs = -2.0f;
    int   bi = base;
#pragma unroll
    for (int k = 0; k < PER_T; ++k) {
        int l = k * P2_THREADS + t;
        float s = s_sc[l];
        if (s > bs) { bs = s; bi = base + l; }
    }
    // wave32 shuffle reduction, tie-break on lowest index
#pragma unroll
    for (int off = 16; off > 0; off >>= 1) {
        float os = __shfl_down(bs, off, 32);
        int   oi = __shfl_down(bi, off, 32);
        if (os > bs || (os == bs && oi < bi)) { bs = os; bi = oi; }
    }
    if ((t & 31) == 0) { w_s[t >> 5] = bs; w_i[t >> 5] = bi; }
    __syncthreads();
    if (t == 0) {
#pragma unroll
        for (int i = 1; i < NWAVES; ++i)
            if (w_s[i] > bs || (w_s[i] == bs && w_i[i] < bi)) {
                bs = w_s[i]; bi = w_i[i];
            }
        cand_s[blockIdx.x] = bs;
        cand_i[blockIdx.x] = bi;
    }
}

// =====================================================================
// Phase 2: persistent 300-step NMS. Each block keeps its 4096-box chunk
// (scores + boxes, 80 KB) resident in LDS; only one 8 B candidate per
// block crosses the device barrier each step.
// =====================================================================
__global__ __launch_bounds__(P2_THREADS)
void nms_phase2(const float* __restrict__ live_in,
                const int* __restrict__ cls,
                const float4* __restrict__ boxesc,
                float* __restrict__ out,       // [1200 box][300 sc][300 cls][300 valid]
                float* __restrict__ cand_s,
                int* __restrict__ cand_i,
                unsigned int* __restrict__ barcnt)
{
    __shared__ float  s_sc[CHUNK];             // 16 KB
    __shared__ float4 s_bx[CHUNK];             // 64 KB
    __shared__ float  w_s[NWAVES];
    __shared__ int    w_i[NWAVES];
    __shared__ float  s_winscore;
    __shared__ int    s_winidx;

    const int t = threadIdx.x;
    const int b = blockIdx.x;
    const int base = b * CHUNK;

    // Stage this block's chunk into LDS (coalesced).
    for (int k = 0; k < PER_T; ++k) {
        int l = k * P2_THREADS + t;
        s_sc[l] = live_in[base + l];
        s_bx[l] = boxesc[base + l];
    }
    __syncthreads();

    unsigned int gen = 0;
    block_argmax_to_cand(s_sc, base, w_s, w_i, cand_s, cand_i);   // prime
    grid_barrier(barcnt, ++gen * NWG);

    for (int it = 0; it < MAX_SEL; ++it) {
        // --- global winner from the 64 per-block candidates (wave 0) ---
        if (t < 32) {
            float bs = cand_s[t];
            int   bi = cand_i[t];
            float o2 = cand_s[t + 32];
            int   i2 = cand_i[t + 32];
            if (o2 > bs || (o2 == bs && i2 < bi)) { bs = o2; bi = i2; }
#pragma unroll
            for (int off = 16; off > 0; off >>= 1) {
                float os = __shfl_down(bs, off, 32);
                int   oi = __shfl_down(bi, off, 32);
                if (os > bs || (os == bs && oi < bi)) { bs = os; bi = oi; }
            }
            if (t == 0) { s_winscore = bs; s_winidx = bi; }
        }
        __syncthreads();

        const float wsc   = s_winscore;
        const int   wi    = s_winidx;
        const bool  valid = (wsc >= 0.0f);
        const float4 wb   = boxesc[wi];        // read-only; uniform index

        if (b == 0 && t == 0) {
            out[it * 4 + 0] = valid ? wb.x : 0.0f;
            out[it * 4 + 1] = valid ? wb.y : 0.0f;
            out[it * 4 + 2] = valid ? wb.z : 0.0f;
            out[it * 4 + 3] = valid ? wb.w : 0.0f;
            out[4 * MAX_SEL + it]            = valid ? wsc : 0.0f;
            out[5 * MAX_SEL + it]            = valid ? (float)cls[wi] : -1.0f;
            out[6 * MAX_SEL + it]            = valid ? 1.0f : 0.0f;
        }

        // --- suppress within this block's LDS-resident chunk ---
        if (valid) {
            const float areaA = fmaxf(wb.z - wb.x, 0.0f) * fmaxf(wb.w - wb.y, 0.0f);
#pragma unroll
            for (int k = 0; k < PER_T; ++k) {
                int l = k * P2_THREADS + t;
                float s = s_sc[l];
                if (s < 0.0f) continue;                 // already dead
                if (base + l == wi) { s_sc[l] = -1.0f; continue; }  // arange==i
                float4 bb = s_bx[l];
                float y1 = fmaxf(wb.x, bb.x);
                float x1 = fmaxf(wb.y, bb.y);
                float y2 = fminf(wb.z, bb.z);
                float x2 = fminf(wb.w, bb.w);
                float inter = fmaxf(y2 - y1, 0.0f) * fmaxf(x2 - x1, 0.0f);
                float areaB = fmaxf(bb.z - bb.x, 0.0f) * fmaxf(bb.w - bb.y, 0.0f);
                float iou   = inter / (areaA + areaB - inter + 1e-9f);
                if (iou > IOU_TH) s_sc[l] = -1.0f;
            }
        }

        // --- next-step local argmax + candidate publish + device barrier ---
        block_argmax_to_cand(s_sc, base, w_s, w_i, cand_s, cand_i);
        grid_barrier(barcnt, ++gen * NWG);
    }
}

// =====================================================================
// Launch
// =====================================================================
extern "C" void kernel_launch(void* const* d_in, const int* in_sizes, int n_in,
                              void* d_out, int out_size, void* d_ws, size_t ws_size,
                              hipStream_t stream)
{
    const float* in = (const float*)d_in[0];
    char* ws = (char*)d_ws;

    // workspace layout (all offsets 16B-aligned where needed)
    float*        live   = (float*)(ws);                         // 1 MB
    float4*       boxesc = (float4*)(ws + (size_t)1048576);      // 4 MB
    int*          cls    = (int*)(ws + (size_t)5242880);         // 1 MB
    float*        cand_s = (float*)(ws + (size_t)6291456);       // 256 B
    int*          cand_i = (int*)(ws + (size_t)6291712);         // 256 B
    unsigned int* barcnt = (unsigned int*)(ws + (size_t)6291968);

    // reset the barrier counter every call (memset node: graph-capture safe)
    (void)hipMemsetAsync(barcnt, 0, 64, stream);

    nms_phase1<<<N_BOXES / P1_ROWS, P1_THREADS, 0, stream>>>(in, live, cls, boxesc);
    nms_phase2<<<NWG, P2_THREADS, 0, stream>>>(live, cls, boxesc,
                                               (float*)d_out, cand_s, cand_i, barcnt);
}